// CaptionDetectionTargetLayer_13640816132819
// MI455X (gfx1250) — compile-verified
//
#include <hip/hip_runtime.h>
#include <hip/hip_bf16.h>
#include <stdint.h>

// ---------------- problem constants (match reference) ----------------
static constexpr int B_IMG  = 16;
static constexpr int N_PROP = 6000;
static constexpr int G_GT   = 300;
static constexpr int L_CAP  = 15;
static constexpr int T_ROI  = 200;
static constexpr int P_MAXC = 66;      // int(T * 0.33)
static constexpr int NBUCK  = 512;     // selection histogram buckets
static constexpr int CMAX   = 768;     // max candidates kept per stream

// ---------------- CDNA5 helpers ----------------
// Low 32 bits of a generic pointer to __shared__ data are the LDS byte offset
// (flat->LDS aperture mapping).
__device__ __forceinline__ uint32_t lds_off32(const void* p) {
  return (uint32_t)(uintptr_t)p;
}

// gfx1250 async global->LDS copy, 16 bytes per lane (ASYNCcnt-tracked).
__device__ __forceinline__ void async_load_b128(uint32_t lds_byte_off, const void* gptr) {
  asm volatile("global_load_async_to_lds_b128 %0, %1, off"
               :: "v"(lds_byte_off), "v"((unsigned long long)(uintptr_t)gptr)
               : "memory");
}

__device__ __forceinline__ void wait_async0() {
#if __has_builtin(__builtin_amdgcn_s_wait_asynccnt)
  __builtin_amdgcn_s_wait_asynccnt(0);
#else
  asm volatile("s_wait_asynccnt 0" ::: "memory");
#endif
}

// Deterministic per-(image, stream, item) uniform in [0,1): splitmix64 hash.
__device__ __forceinline__ float urand(uint32_t img, uint32_t idx, uint32_t salt) {
  uint64_t x = ((uint64_t)img << 48) | ((uint64_t)salt << 32) | (uint64_t)idx;
  x += 0x9E3779B97F4A7C15ull;
  x ^= x >> 30; x *= 0xBF58476D1CE4E5B9ull;
  x ^= x >> 27; x *= 0x94D049BB133111EBull;
  x ^= x >> 31;
  return (float)((uint32_t)x >> 8) * (1.0f / 16777216.0f);
}

__device__ __forceinline__ int bucket_of(float k) {
  int q = (int)(k * (float)NBUCK);
  return (q < NBUCK - 1) ? q : (NBUCK - 1);
}

// ---------------- Kernel 1: IoU max/argmax + masked shuffle keys ----------------
__global__ void __launch_bounds__(256)
iou_stage(const float* __restrict__ props, const float* __restrict__ gtb,
          int* __restrict__ amax, float* __restrict__ key_pos,
          float* __restrict__ key_neg) {
  __shared__ float sgt[G_GT * 4];
  __shared__ float sarea[G_GT];
  __shared__ int   svalid[G_GT];

  const int b   = blockIdx.y;
  const int tid = threadIdx.x;

  // Async-stage this image's GT boxes into LDS (300 x 16B).
  for (int g = tid; g < G_GT; g += blockDim.x)
    async_load_b128(lds_off32(&sgt[g * 4]), &gtb[(size_t)(b * G_GT + g) * 4]);
  wait_async0();
  __syncthreads();

  for (int g = tid; g < G_GT; g += blockDim.x) {
    float y1 = sgt[g*4+0], x1 = sgt[g*4+1], y2 = sgt[g*4+2], x2 = sgt[g*4+3];
    svalid[g] = (fabsf(y1) + fabsf(x1) + fabsf(y2) + fabsf(x2)) > 0.0f;
    sarea[g]  = (y2 - y1) * (x2 - x1);
  }
  __syncthreads();

  const int i = blockIdx.x * blockDim.x + tid;
  if (i >= N_PROP) return;

  const float* p = &props[(size_t)(b * N_PROP + i) * 4];
  float py1 = p[0], px1 = p[1], py2 = p[2], px2 = p[3];
  bool  pv    = (fabsf(py1) + fabsf(px1) + fabsf(py2) + fabsf(px2)) > 0.0f;
  float parea = (py2 - py1) * (px2 - px1);

  float best = -1.0f; int besta = 0;
  for (int g = 0; g < G_GT; ++g) {
    if (!svalid[g]) continue;
    float iy1 = fmaxf(py1, sgt[g*4+0]);
    float ix1 = fmaxf(px1, sgt[g*4+1]);
    float iy2 = fminf(py2, sgt[g*4+2]);
    float ix2 = fminf(px2, sgt[g*4+3]);
    float inter = fmaxf(iy2 - iy1, 0.0f) * fmaxf(ix2 - ix1, 0.0f);
    float uni   = parea + sarea[g] - inter;
    float iou   = (uni > 0.0f) ? inter / fmaxf(uni, 1e-12f) : 0.0f;
    if (iou > best) { best = iou; besta = g; }
  }

  bool pos = (best >= 0.5f) && pv;
  bool neg = (best <  0.5f) && pv;
  const int o = b * N_PROP + i;
  amax[o]    = besta;
  key_pos[o] = pos ? urand((uint32_t)b, (uint32_t)i, 0u) : 1e9f;
  key_neg[o] = neg ? urand((uint32_t)b, (uint32_t)i, 1u) : 1e9f;
}

// ---------------- Kernel 2: bucket-select samples + emit outputs ----------------
__global__ void __launch_bounds__(1024)
select_emit(const float* __restrict__ props, const float* __restrict__ gtb,
            const int* __restrict__ gtc, const float* __restrict__ gts,
            const int* __restrict__ amax, const float* __restrict__ key_pos,
            const float* __restrict__ key_neg, float* __restrict__ out) {
  __shared__ float skey[2][N_PROP];     // staged keys (pos, neg)      48000 B
  __shared__ int   hist[2][NBUCK];      // selection histograms         4096 B
  __shared__ int   cand[2][CMAX];       // candidate item indices       6144 B
  __shared__ int   sel[2][T_ROI];       // final orderings              1600 B
  __shared__ int   ccnt[2];             // candidate counts
  __shared__ int   cutb[2];             // cutoff bucket per stream
  __shared__ int   numt[2];             // num_pos, num_neg
  __shared__ int   gvalid_s;

  const int b    = blockIdx.x;
  const int tid  = threadIdx.x;
  const int nthr = blockDim.x;

  // --- init LDS state ---
  for (int k = tid; k < 2 * NBUCK; k += nthr) (&hist[0][0])[k] = 0;
  for (int k = tid; k < 2 * T_ROI; k += nthr) (&sel[0][0])[k] = 0;
  if (tid < 2) ccnt[tid] = 0;
  if (tid == 0) gvalid_s = 0;

  // --- async-stage both key arrays into LDS (2 x 24KB, 16B chunks) ---
  for (int c = tid; c < N_PROP / 4; c += nthr) {
    async_load_b128(lds_off32(&skey[0][c * 4]), &key_pos[(size_t)b * N_PROP + c * 4]);
    async_load_b128(lds_off32(&skey[1][c * 4]), &key_neg[(size_t)b * N_PROP + c * 4]);
  }
  wait_async0();
  __syncthreads();

  // --- bucket histogram pass + valid-GT count ---
  for (int i = tid; i < N_PROP; i += nthr) {
    float kp = skey[0][i];
    if (kp < 1e9f) atomicAdd(&hist[0][bucket_of(kp)], 1);
    float kn = skey[1][i];
    if (kn < 1e9f) atomicAdd(&hist[1][bucket_of(kn)], 1);
  }
  if (tid < G_GT) {
    const float* g = &gtb[(size_t)(b * G_GT + tid) * 4];
    if (fabsf(g[0]) + fabsf(g[1]) + fabsf(g[2]) + fabsf(g[3]) > 0.0f)
      atomicAdd(&gvalid_s, 1);
  }
  __syncthreads();

  // --- cutoff bucket so that cum count >= min(T, total); also totals ---
  if (tid < 2) {
    int total = 0;
#pragma unroll 8
    for (int q = 0; q < NBUCK; ++q) total += hist[tid][q];
    int want = (total < T_ROI) ? total : T_ROI;
    int cum = 0, cb = NBUCK - 1;
    for (int q = 0; q < NBUCK; ++q) {
      cum += hist[tid][q];
      if (cum >= want) { cb = q; break; }
    }
    cutb[tid] = cb;
    numt[tid] = total;
  }
  __syncthreads();

  // --- compact candidates (bucket <= cutoff) ---
  for (int i = tid; i < N_PROP; i += nthr) {
    float kp = skey[0][i];
    if (kp < 1e9f && bucket_of(kp) <= cutb[0]) {
      int s = atomicAdd(&ccnt[0], 1);
      if (s < CMAX) cand[0][s] = i;
    }
    float kn = skey[1][i];
    if (kn < 1e9f && bucket_of(kn) <= cutb[1]) {
      int s = atomicAdd(&ccnt[1], 1);
      if (s < CMAX) cand[1][s] = i;
    }
  }
  __syncthreads();

  // --- exact (key, idx) rank among candidates == global rank ---
  for (int s = 0; s < 2; ++s) {
    const int C = (ccnt[s] < CMAX) ? ccnt[s] : CMAX;
    for (int c = tid; c < C; c += nthr) {
      const int   i  = cand[s][c];
      const float ki = skey[s][i];
      int r = 0;
      for (int j = 0; j < C; ++j) {
        const int   ij = cand[s][j];
        const float kj = skey[s][ij];
        r += (kj < ki) || (kj == ki && ij < i);
      }
      if (r < T_ROI) sel[s][r] = i;
    }
  }
  __syncthreads();

  const int num_pos = numt[0], num_neg = numt[1], g_valid = gvalid_s;
  const int pos_count = (num_pos < P_MAXC) ? num_pos : P_MAXC;
  int neg_count = (int)((1.0f / 0.33f) * (float)pos_count) - pos_count;
  neg_count = max(0, min(neg_count, num_neg));

  // Output layout: rois[B,T,4] | deltas[B,T,4] | caps[B,T,L] | scores[B,T]
  float* o_rois = out;
  float* o_del  = out + (size_t)B_IMG * T_ROI * 4;
  float* o_caps = out + (size_t)2 * B_IMG * T_ROI * 4;
  float* o_sc   = out + (size_t)2 * B_IMG * T_ROI * 4 + (size_t)B_IMG * T_ROI * L_CAP;

  for (int j = tid; j < T_ROI; j += nthr) {
    float vr[4] = {0, 0, 0, 0}, vd[4] = {0, 0, 0, 0};
    float capv[L_CAP];
#pragma unroll
    for (int l = 0; l < L_CAP; ++l) capv[l] = 0.0f;
    float sc = -1.0f;

    if (j < pos_count) {
      const int idx = sel[0][j];
      const float* p = &props[(size_t)(b * N_PROP + idx) * 4];
      vr[0] = p[0]; vr[1] = p[1]; vr[2] = p[2]; vr[3] = p[3];
      const int a = amax[b * N_PROP + idx];
      const float* g = &gtb[(size_t)(b * G_GT + a) * 4];
      float h  = vr[2] - vr[0], w  = vr[3] - vr[1];
      float cy = vr[0] + 0.5f * h, cx = vr[1] + 0.5f * w;
      float gh = g[2] - g[0],  gw = g[3] - g[1];
      float gcy = g[0] + 0.5f * gh, gcx = g[1] + 0.5f * gw;
      float hs  = (h  > 0.0f) ? h  : 1.0f;
      float wsx = (w  > 0.0f) ? w  : 1.0f;
      float ghs = (gh > 0.0f) ? gh : 1.0f;
      float gws = (gw > 0.0f) ? gw : 1.0f;
      vd[0] = ((gcy - cy) / hs)  / 0.1f;
      vd[1] = ((gcx - cx) / wsx) / 0.1f;
      vd[2] = logf(ghs / hs)     / 0.2f;
      vd[3] = logf(gws / wsx)    / 0.2f;
#pragma unroll
      for (int l = 0; l < L_CAP; ++l)
        capv[l] = (float)gtc[((size_t)(b * G_GT + a)) * L_CAP + l];
      int gm = max(g_valid - 1, 0);
      int sidx = min(min(idx, gm), G_GT - 1);
      sc = gts[b * G_GT + sidx];
    } else if (j < pos_count + neg_count) {
      const int idx = sel[1][j - pos_count];
      const float* p = &props[(size_t)(b * N_PROP + idx) * 4];
      vr[0] = p[0]; vr[1] = p[1]; vr[2] = p[2]; vr[3] = p[3];
      sc = 0.0f;
    }

#pragma unroll
    for (int k = 0; k < 4; ++k) {
      o_rois[((size_t)(b * T_ROI + j)) * 4 + k] = vr[k];
      o_del [((size_t)(b * T_ROI + j)) * 4 + k] = vd[k];
    }
#pragma unroll
    for (int l = 0; l < L_CAP; ++l)
      o_caps[((size_t)(b * T_ROI + j)) * L_CAP + l] = capv[l];
    o_sc[b * T_ROI + j] = sc;
  }
}

// ---------------- host launch ----------------
extern "C" void kernel_launch(void* const* d_in, const int* in_sizes, int n_in,
                              void* d_out, int out_size, void* d_ws, size_t ws_size,
                              hipStream_t stream) {
  (void)in_sizes; (void)n_in; (void)out_size; (void)ws_size;
  const float* props = (const float*)d_in[0];   // [16,6000,4]
  const float* gtb   = (const float*)d_in[1];   // [16,300,4]
  const int*   gtc   = (const int*)  d_in[2];   // [16,300,15]
  const float* gts   = (const float*)d_in[3];   // [16,300]

  // Workspace: argmax[B*N] int | key_pos[B*N] f32 | key_neg[B*N] f32
  char* ws = (char*)d_ws;
  int*   amax    = (int*)  (ws);
  float* key_pos = (float*)(ws + (size_t)B_IMG * N_PROP * 4);
  float* key_neg = (float*)(ws + (size_t)2 * B_IMG * N_PROP * 4);

  dim3 g1((N_PROP + 255) / 256, B_IMG);
  iou_stage<<<g1, 256, 0, stream>>>(props, gtb, amax, key_pos, key_neg);
  select_emit<<<B_IMG, 1024, 0, stream>>>(props, gtb, gtc, gts, amax,
                                          key_pos, key_neg, (float*)d_out);
}